// DifferentiableSuperpixelTokenizer_86028194939428
// MI455X (gfx1250) — compile-verified
//
#include <hip/hip_runtime.h>

// Problem constants from the reference (fixed shapes).
constexpr int B  = 16;
constexpr int C  = 192;
constexpr int H  = 224;
constexpr int W  = 224;
constexpr int MS = 196;          // max_segments
constexpr int HW = H * W;        // 50176

constexpr int THREADS   = 256;   // 8 waves (wave32)
constexpr int TILE_P    = 3136;  // pixels per scatter block (HW / 16)
constexpr int BLK_PER_B = HW / TILE_P;       // 16 blocks per batch
constexpr int CH_GRP    = 48;                // channels resident in LDS per pass
constexpr int NPASS     = C / CH_GRP;        // 4

typedef __attribute__((ext_vector_type(2))) float v2f;
typedef __attribute__((ext_vector_type(8))) float v8f;

// ---------------------------------------------------------------------------
// Kernel 1: zero the workspace accumulators (sums + counts).
// ---------------------------------------------------------------------------
__global__ void zero_ws_kernel(float* __restrict__ ws, int n) {
  int i = blockIdx.x * blockDim.x + threadIdx.x;
  int stride = gridDim.x * blockDim.x;
  for (; i < n; i += stride) ws[i] = 0.0f;
}

// ---------------------------------------------------------------------------
// Kernel 2: two-stage segment scatter-add.
// One block = (batch b, tile of TILE_P pixels). Segment IDs are cached in LDS
// once (int4 loads in, ds_load_b128 out); per-segment partial sums for CH_GRP
// channels accumulate in LDS via ds float atomics, then flush to global with
// coalesced global atomics. Features (616 MB) are read exactly once, as
// float4 -> global_load_b128 to keep the VMEM instruction rate 4x down.
// ---------------------------------------------------------------------------
__global__ __launch_bounds__(THREADS)
void scatter_kernel(const float* __restrict__ feat,
                    const int*   __restrict__ seg,
                    float* __restrict__ gsum,
                    float* __restrict__ gcnt) {
  __shared__ float lsum[CH_GRP * MS];  // [c_local][segment] -> bank-friendly
  __shared__ float lcnt[MS];
  __shared__ int   lseg[TILE_P];

  const int tid = threadIdx.x;
  const int b   = blockIdx.x / BLK_PER_B;
  const int p0  = (blockIdx.x % BLK_PER_B) * TILE_P;

  for (int i = tid; i < MS; i += THREADS) lcnt[i] = 0.0f;
  __syncthreads();

  // Cache this tile's segment ids in LDS (vectorized) and count pixels/seg.
  {
    const int4* segp4 = reinterpret_cast<const int4*>(seg + (size_t)b * HW + p0);
    int4* lseg4 = reinterpret_cast<int4*>(lseg);
    for (int p4 = tid; p4 < TILE_P / 4; p4 += THREADS) {
      int4 s = segp4[p4];
      lseg4[p4] = s;
      __hip_atomic_fetch_add(&lcnt[s.x], 1.0f, __ATOMIC_RELAXED,
                             __HIP_MEMORY_SCOPE_WORKGROUP);
      __hip_atomic_fetch_add(&lcnt[s.y], 1.0f, __ATOMIC_RELAXED,
                             __HIP_MEMORY_SCOPE_WORKGROUP);
      __hip_atomic_fetch_add(&lcnt[s.z], 1.0f, __ATOMIC_RELAXED,
                             __HIP_MEMORY_SCOPE_WORKGROUP);
      __hip_atomic_fetch_add(&lcnt[s.w], 1.0f, __ATOMIC_RELAXED,
                             __HIP_MEMORY_SCOPE_WORKGROUP);
    }
  }
  __syncthreads();

  const int4* lseg4 = reinterpret_cast<const int4*>(lseg);

  for (int pass = 0; pass < NPASS; ++pass) {
    const int c0 = pass * CH_GRP;

    for (int i = tid; i < CH_GRP * MS; i += THREADS) lsum[i] = 0.0f;
    __syncthreads();

    // Accumulate: float4 per thread -> global_load_b128, coalesced along W.
    for (int cl = 0; cl < CH_GRP; ++cl) {
      const float4* fp4 = reinterpret_cast<const float4*>(
          feat + ((size_t)b * C + (size_t)(c0 + cl)) * HW + p0);
      float* row = &lsum[cl * MS];
      for (int p4 = tid; p4 < TILE_P / 4; p4 += THREADS) {
        float4 f = fp4[p4];
        int4   s = lseg4[p4];     // ds_load_b128 from the cached seg tile
        __hip_atomic_fetch_add(&row[s.x], f.x, __ATOMIC_RELAXED,
                               __HIP_MEMORY_SCOPE_WORKGROUP);
        __hip_atomic_fetch_add(&row[s.y], f.y, __ATOMIC_RELAXED,
                               __HIP_MEMORY_SCOPE_WORKGROUP);
        __hip_atomic_fetch_add(&row[s.z], f.z, __ATOMIC_RELAXED,
                               __HIP_MEMORY_SCOPE_WORKGROUP);
        __hip_atomic_fetch_add(&row[s.w], f.w, __ATOMIC_RELAXED,
                               __HIP_MEMORY_SCOPE_WORKGROUP);
      }
    }
    __syncthreads();

    // Flush partials: consecutive tid -> consecutive channel so the global
    // float atomics are coalesced along C.
    for (int i = tid; i < CH_GRP * MS; i += THREADS) {
      int cl = i % CH_GRP;
      int s  = i / CH_GRP;
      atomicAdd(&gsum[((size_t)b * MS + s) * C + (c0 + cl)], lsum[cl * MS + s]);
    }
    if (pass == 0) {
      for (int s = tid; s < MS; s += THREADS)
        atomicAdd(&gcnt[b * MS + s], lcnt[s]);
    }
    __syncthreads();
  }
}

// ---------------------------------------------------------------------------
// Kernel 3: fused finalize. One wave computes one 16x16 tile of the
// positional-embedding GEMM with V_WMMA_F32_16X16X4_F32 (K=2 zero-padded to
// 4), then fuses mean = sum * rcp(max(cnt,1)), +bias, +pos and stores.
//   M = B*MS = 3136 -> 196 M-tiles, N = C = 192 -> 12 N-tiles, 2352 tiles.
// ---------------------------------------------------------------------------
constexpr int NT    = C / 16;            // 12
constexpr int MT    = (B * MS) / 16;     // 196
constexpr int TILES = NT * MT;           // 2352
constexpr int WPB   = THREADS / 32;      // 8 waves per block

__global__ __launch_bounds__(THREADS)
void finalize_kernel(const float* __restrict__ gsum,
                     const float* __restrict__ gcnt,
                     const float* __restrict__ cen,    // (B, MS, 2)
                     const float* __restrict__ posW,   // (2, C)
                     const float* __restrict__ posb,   // (C)
                     float* __restrict__ out) {        // (B, MS, C)
  const int lane = threadIdx.x & 31;
  const int wave = blockIdx.x * WPB + (threadIdx.x >> 5);  // < TILES exactly
  const int tm   = wave / NT;
  const int tn   = wave % NT;
  const int hi   = lane & 16;   // lanes 16..31 hold K=2,3 (zero pad)
  const int l16  = lane & 15;

  // A (16x4 f32): lane m=l16; VGPR0=K0, VGPR1=K1 for lanes<16; zeros above.
  // B (4x16 f32): lane n=l16; VGPR0=K0 row, VGPR1=K1 row; zeros above.
  v2f a;  a.x = 0.0f;  a.y = 0.0f;
  v2f bm; bm.x = 0.0f; bm.y = 0.0f;
  if (!hi) {
    int r    = tm * 16 + l16;          // global token row in [0, B*MS)
    int bi   = r / MS;
    int s    = r - bi * MS;
    const float* cp = cen + ((size_t)bi * MS + s) * 2;
    a.x = cp[0] * (1.0f / (float)W);   // scaled centroid x  (K=0)
    a.y = cp[1] * (1.0f / (float)H);   // scaled centroid y  (K=1)
    int c = tn * 16 + l16;
    bm.x = posW[c];                    // pos_W[0][c]        (K=0)
    bm.y = posW[C + c];                // pos_W[1][c]        (K=1)
  }

  v8f acc = {};
  // 8 args: (neg_a, A, neg_b, B, c_mod, C, reuse_a, reuse_b)
  acc = __builtin_amdgcn_wmma_f32_16x16x4_f32(
      false, a, false, bm, (short)0, acc, false, false);

  // D layout: VGPR v, lanes 0-15 -> M=v, lanes 16-31 -> M=v+8; N = l16.
  const int   col   = tn * 16 + l16;
  const float pb    = posb[col];
  const int   rbase = tm * 16 + (hi ? 8 : 0);
#pragma unroll
  for (int v = 0; v < 8; ++v) {
    int    row = rbase + v;
    size_t idx = (size_t)row * C + col;
    // Counts are integers >= 1 after the max; a single v_rcp_f32 (~1 ulp) is
    // ample precision for a mean -- avoids the full IEEE div expansion.
    float  inv = __builtin_amdgcn_rcpf(fmaxf(gcnt[row], 1.0f));
    out[idx] = acc[v] + pb + gsum[idx] * inv;
  }
}

// ---------------------------------------------------------------------------
// Host launcher. Inputs (setup_inputs order):
//   0: img (unused)  1: features  2: segments  3: centroid_coords
//   4: pos_W  5: pos_b  6: max_segments (scalar)
// ---------------------------------------------------------------------------
extern "C" void kernel_launch(void* const* d_in, const int* in_sizes, int n_in,
                              void* d_out, int out_size, void* d_ws,
                              size_t ws_size, hipStream_t stream) {
  const float* feat = (const float*)d_in[1];
  const int*   seg  = (const int*)d_in[2];
  const float* cen  = (const float*)d_in[3];
  const float* posW = (const float*)d_in[4];
  const float* posb = (const float*)d_in[5];

  float* gsum = (float*)d_ws;                       // B*MS*C floats
  float* gcnt = gsum + (size_t)B * MS * C;          // B*MS floats

  const int nzero = B * MS * C + B * MS;            // 605,248 floats
  zero_ws_kernel<<<(nzero + THREADS - 1) / THREADS, THREADS, 0, stream>>>(
      gsum, nzero);

  scatter_kernel<<<B * BLK_PER_B, THREADS, 0, stream>>>(feat, seg, gsum, gcnt);

  finalize_kernel<<<TILES / WPB, THREADS, 0, stream>>>(
      gsum, gcnt, cen, posW, posb, (float*)d_out);
}